// NeighbourSparseAttention_80161269612942
// MI455X (gfx1250) — compile-verified
//
#include <hip/hip_runtime.h>

// MI455X (gfx1250) NeighbourSparseAttention
//
// Exact algebraic refactoring:
//   local[i] = node[i] . (W_emb @ w_local) + (b_emb.w_local + b_local)
//   neigh[i] = node[i] . (W_emb @ w_neigh) + (b_emb.w_neigh + b_neigh)
//   out[i]   = local[i] + sum_{e: src==i} neigh[dst[e]]
//
// Roofline: one streaming pass over node_tensor (204.8 MB -> ~8.8us @ 23.3
// TB/s) dominates. The [N,512]@[512,2] projection runs as WMMA
// V_WMMA_F32_16X16X4_F32 (full fp32). A-tiles are staged LDS-side by the
// Tensor Data Mover (1 descriptor per 16x256 half-tile, double buffered via
// TENSORcnt), with TDM pad_interval=128dw/pad_amount=2dw so the per-lane
// ds_load_b64 feed is 64-bank conflict-free.

typedef __attribute__((ext_vector_type(2))) float v2f;
typedef __attribute__((ext_vector_type(8))) float v8f;
typedef __attribute__((ext_vector_type(4))) unsigned int v4u;
typedef __attribute__((ext_vector_type(8))) int v8i;
typedef __attribute__((ext_vector_type(4))) int v4i;

#define NSA_SIZE 512
#define WAVES 4           // waves (tiles) per block
#define RSTRIDE 260       // padded LDS row stride in dwords (256 + 2 pads * 2dw)
#define HALF_DW (16 * RSTRIDE)  // dwords per 16x256 half-tile buffer

// ---------------- K1: fold the 512x512 GEMM into two 512-vectors ------------
__global__ void nsa_precompute(const float* __restrict__ W,
                               const float* __restrict__ bemb,
                               const float* __restrict__ wl,
                               const float* __restrict__ bl,
                               const float* __restrict__ wn,
                               const float* __restrict__ bn,
                               float* __restrict__ vloc,
                               float* __restrict__ vngh,
                               float* __restrict__ cons) {
  __shared__ float s0[256];
  __shared__ float s1[256];
  const int b = blockIdx.x, t = threadIdx.x;
  float p0 = 0.f, p1 = 0.f;
  if (b < NSA_SIZE) {
    const float* row = W + (size_t)b * NSA_SIZE;
    for (int j = t; j < NSA_SIZE; j += 256) {
      float r = row[j];
      p0 += r * wl[j];
      p1 += r * wn[j];
    }
  } else {
    for (int j = t; j < NSA_SIZE; j += 256) {
      float r = bemb[j];
      p0 += r * wl[j];
      p1 += r * wn[j];
    }
  }
  s0[t] = p0; s1[t] = p1;
  __syncthreads();
  for (int s = 128; s > 0; s >>= 1) {
    if (t < s) { s0[t] += s0[t + s]; s1[t] += s1[t + s]; }
    __syncthreads();
  }
  if (t == 0) {
    if (b < NSA_SIZE) { vloc[b] = s0[0]; vngh[b] = s1[0]; }
    else              { cons[0] = s0[0] + bl[0]; cons[1] = s1[0] + bn[0]; }
  }
}

// --------- TDM: one D# describing a 16-row x 256-float tile -> LDS ----------
// D# group0: {count=1, lds_addr, global_addr(57b), type=2}
// D# group1: data_size=4B(2), pad_enable, pad_interval=6 (128 dw),
//            pad_amount=1 (2 dw) -> LDS row stride 260 dw;
//            tensor_dim0=512, tensor_dim1=rows_avail (HW zero-fill OOB rows),
//            tile_dim0=256, tile_dim1=16, tensor_dim0_stride=512.
__device__ __forceinline__ void tdm_load_half(unsigned long long ga,
                                              unsigned lds, int rows_avail) {
  v4u g0;
  g0[0] = 1u;                                        // count=1, user mode
  g0[1] = lds;                                       // lds_addr (bytes)
  g0[2] = (unsigned)ga;                              // global_addr[31:0]
  g0[3] = (unsigned)((ga >> 32) & 0x01FFFFFFu) | 0x80000000u;  // [56:32]|type=2
  v8i g1;
  g1[0] = (2 << 16) | (1 << 20) | (6 << 22) | (1 << 25); // 4B, pad 2dw/128dw
  g1[1] = (int)((NSA_SIZE & 0xffff) << 16);          // tensor_dim0=512 lo16
  g1[2] = (int)(((unsigned)rows_avail) << 16);       // dim0 hi=0 | tensor_dim1 lo16
  g1[3] = (int)(256u << 16);                         // dim1 hi=0 | tile_dim0=256
  g1[4] = 16;                                        // tile_dim1=16, tile_dim2=0
  g1[5] = NSA_SIZE;                                  // tensor_dim0_stride lo32
  g1[6] = 0;
  g1[7] = 0;
  v4i gz = {0, 0, 0, 0};                             // 2-D: groups 2/3 unused
#if __has_include(<hip/amd_detail/amd_gfx1250_TDM.h>)
  v8i gz8 = {0, 0, 0, 0, 0, 0, 0, 0};
  __builtin_amdgcn_tensor_load_to_lds(g0, g1, gz, gz, gz8, 0);
#else
  __builtin_amdgcn_tensor_load_to_lds(g0, g1, gz, gz, 0);
#endif
}

// ------------- K2: [N,512] @ [512, {v_local,v_neigh,0...}] via WMMA ---------
// One wave per 16-row tile; two TDM half-tiles double-buffered on TENSORcnt.
// A 16x4 f32 lane layout: lane l (m=l&15, kh=l>>4) holds A[m][k+2kh..+1].
// LDS dword of element (m,k'): m*260 + k' + 2*(k'>>7)  (TDM pad layout).
__global__ void __launch_bounds__(WAVES * 32)
nsa_gemv_wmma(const float* __restrict__ node,
              const float* __restrict__ vloc,
              const float* __restrict__ vngh,
              const float* __restrict__ cons,
              float* __restrict__ out_local,
              float* __restrict__ ngh_out,
              int Nn) {
  __shared__ float smem[WAVES * 2 * HALF_DW];        // 133,120 bytes

  const int lane = threadIdx.x & 31;
  const int wave = threadIdx.x >> 5;
  const int tile = blockIdx.x * WAVES + wave;
  const int n    = lane & 15;   // M for A / N for B,C
  const int kh   = lane >> 4;   // K half-pair carried by this lane
  const int row0 = tile * 16;

  int ra = Nn - row0;
  ra = ra < 0 ? 0 : (ra > 16 ? 16 : ra);             // rows available (HW zero-fills rest)

  float* wbuf = smem + (size_t)wave * (2 * HALF_DW);
  unsigned lds0 = (unsigned)(size_t)wbuf;            // LDS byte offset of buffer 0
  unsigned lds1 = lds0 + HALF_DW * 4;
  unsigned long long ga =
      (unsigned long long)(size_t)node +
      (unsigned long long)(unsigned)row0 * (NSA_SIZE * 4);

  tdm_load_half(ga, lds0, ra);                       // K = 0..255
  tdm_load_half(ga + 256 * 4, lds1, ra);             // K = 256..511

  const float* bsrc = (n == 0) ? vloc : vngh;
  const float cl = cons[0];
  const float cn = cons[1];
  v8f acc = {};

  // ---- half 0 ----
  __builtin_amdgcn_s_wait_tensorcnt((short)1);       // buffer 0 landed
  asm volatile("" ::: "memory");
  {
    const float* ap = wbuf + n * RSTRIDE + 2 * kh;
#pragma unroll 8
    for (int k = 0; k < 256; k += 4) {
      v2f a = *(const v2f*)(ap + k + 2 * (k >> 7));  // conflict-free ds_load_b64
      v2f b = {0.f, 0.f};
      if (n < 2) b = *(const v2f*)(bsrc + k + 2 * kh);
      acc = __builtin_amdgcn_wmma_f32_16x16x4_f32(
          false, a, false, b, (short)0, acc, false, false);
    }
  }
  // ---- half 1 ----
  __builtin_amdgcn_s_wait_tensorcnt((short)0);       // buffer 1 landed
  asm volatile("" ::: "memory");
  {
    const float* ap = wbuf + HALF_DW + n * RSTRIDE + 2 * kh;
#pragma unroll 8
    for (int k = 0; k < 256; k += 4) {
      v2f a = *(const v2f*)(ap + k + 2 * (k >> 7));
      v2f b = {0.f, 0.f};
      if (n < 2) b = *(const v2f*)(bsrc + 256 + k + 2 * kh);
      acc = __builtin_amdgcn_wmma_f32_16x16x4_f32(
          false, a, false, b, (short)0, acc, false, false);
    }
  }

  // C/D 16x16 f32 layout: vgpr r, lanes 0-15 -> D[r][n]; lanes 16-31 -> D[8+r][n].
  // Column 0 = local (lanes 0/16), column 1 = neigh (lanes 1/17).
  if (n < 2) {
    const float addc = (n == 0) ? cl : cn;
    float* dst = (n == 0) ? out_local : ngh_out;
    const int rb = row0 + kh * 8;
#pragma unroll
    for (int r = 0; r < 8; ++r) {
      int rr = rb + r;
      if (rr < Nn) dst[rr] = acc[r] + addc;
    }
  }
}

// ---------------- K3: edge scatter (segment_sum) ----------------------------
__global__ void nsa_edges(const int* __restrict__ esrc,
                          const int* __restrict__ edst,
                          const float* __restrict__ ngh,
                          float* __restrict__ out,
                          int E) {
  int i = (blockIdx.x * blockDim.x + threadIdx.x) * 4;
  if (i + 4 <= E) {
    int4 s = *(const int4*)(esrc + i);
    int4 d = *(const int4*)(edst + i);
    unsafeAtomicAdd(&out[s.x], ngh[d.x]);             // gather + atomics are
    unsafeAtomicAdd(&out[s.y], ngh[d.y]);             // L2-resident (400KB)
    unsafeAtomicAdd(&out[s.z], ngh[d.z]);
    unsafeAtomicAdd(&out[s.w], ngh[d.w]);
  } else {
    for (; i < E; ++i) unsafeAtomicAdd(&out[esrc[i]], ngh[edst[i]]);
  }
}

extern "C" void kernel_launch(void* const* d_in, const int* in_sizes, int n_in,
                              void* d_out, int out_size, void* d_ws, size_t ws_size,
                              hipStream_t stream) {
  const float* node = (const float*)d_in[0];
  const int*   esrc = (const int*)d_in[1];
  const int*   edst = (const int*)d_in[2];
  const float* W    = (const float*)d_in[3];
  const float* bemb = (const float*)d_in[4];
  const float* wl   = (const float*)d_in[5];
  const float* bl   = (const float*)d_in[6];
  const float* wn   = (const float*)d_in[7];
  const float* bn   = (const float*)d_in[8];

  const int Nn = in_sizes[0] / NSA_SIZE;
  const int E  = in_sizes[1];

  float* fws  = (float*)d_ws;
  float* vloc = fws;                 // 512 floats
  float* vngh = fws + NSA_SIZE;      // 512 floats
  float* cons = fws + 2 * NSA_SIZE;  // 2 floats
  float* ngh  = fws + 1536;          // N floats (64B-aligned region)
  float* out  = (float*)d_out;

  nsa_precompute<<<NSA_SIZE + 1, 256, 0, stream>>>(W, bemb, wl, bl, wn, bn,
                                                   vloc, vngh, cons);

  const int tiles  = (Nn + 15) / 16;
  const int blocks = (tiles + WAVES - 1) / WAVES;
  nsa_gemv_wmma<<<blocks, WAVES * 32, 0, stream>>>(node, vloc, vngh, cons,
                                                   out, ngh, Nn);

  const int ethreads = (E + 3) / 4;
  nsa_edges<<<(ethreads + 255) / 256, 256, 0, stream>>>(esrc, edst, ngh, out, E);
}